// Ts_corr_48902497632338
// MI455X (gfx1250) — compile-verified
//
#include <hip/hip_runtime.h>
#include <math.h>

typedef float v2f __attribute__((ext_vector_type(2)));
typedef float v8f __attribute__((ext_vector_type(8)));

#define NB    2048   // batches
#define NR    64     // rows
#define NT    30     // time steps
#define WND   10     // window
#define NWIN  21     // NT - WND + 1
#define NPAIR 2080   // 64*65/2
#define XS    33     // padded LDS row stride (odd -> conflict-free strided reads)
#define THREADS 320  // 10 wave32s, one per triu 16x16 tile

__launch_bounds__(THREADS, 2)
__global__ void ts_corr_kernel(const float* __restrict__ x, float* __restrict__ out) {
    __shared__ float xn[NR * XS];       // normalized rows, cols 30..32 zeroed
    __shared__ float rinv[NR * NWIN];   // 1/sqrt(slidmean(xn^2)+1e-8) per row/window

    const int b   = blockIdx.x;
    const int tid = threadIdx.x;
    const float* xb = x + (size_t)b * (NR * NT);

    // ---- Phase 1: cooperative load (coalesced) into padded LDS ----
    for (int idx = tid; idx < NR * NT; idx += THREADS) {
        int r = idx / NT;
        int c = idx - r * NT;
        xn[r * XS + c] = xb[idx];
    }
    __syncthreads();

    // ---- Phase 2: per-row normalize + sliding inverse std (threads 0..63) ----
    if (tid < NR) {
        float* row = &xn[tid * XS];
        float s = 0.f;
        #pragma unroll
        for (int k = 0; k < NT; ++k) s += row[k];
        float mean = s * (1.0f / NT);
        float var = 0.f;
        #pragma unroll
        for (int k = 0; k < NT; ++k) { float d = row[k] - mean; var += d * d; }
        float rstd = 1.0f / sqrtf(var * (1.0f / NT));
        float q[NT];
        #pragma unroll
        for (int k = 0; k < NT; ++k) { q[k] = (row[k] - mean) * rstd; row[k] = q[k]; }
        row[30] = 0.f; row[31] = 0.f; row[32] = 0.f;   // K-padding: dead lanes read here

        float ss = 0.f;
        #pragma unroll
        for (int k = 0; k < WND; ++k) ss += q[k] * q[k];
        rinv[tid * NWIN + 0] = 1.0f / sqrtf(ss * (1.0f / WND) + 1e-8f);
        #pragma unroll
        for (int t = 1; t < NWIN; ++t) {
            ss += q[t + WND - 1] * q[t + WND - 1] - q[t - 1] * q[t - 1];
            rinv[tid * NWIN + t] = 1.0f / sqrtf(ss * (1.0f / WND) + 1e-8f);
        }
    }
    __syncthreads();

    // ---- Phase 3: windowed Gram tiles via f32 WMMA, one wave per triu tile ----
    // Scalarize the wave id so tile indices + diag test are uniform (SGPR) values.
    const int wave = __builtin_amdgcn_readfirstlane(tid >> 5);
    const int lane = tid & 31;
    const int hi   = lane >> 4;    // 0: lanes 0-15 (K=0,1 / M=v), 1: lanes 16-31 (K=2,3 / M=v+8)
    const int lx   = lane & 15;

    // nibble tables: TI = {0,0,0,0,1,1,1,2,2,3}, TJ = {0,1,2,3,1,2,3,2,3,3}
    const int ti = (int)((0x3221110000ULL >> (4 * wave)) & 0xF);
    const int tj = (int)((0x3323213210ULL >> (4 * wave)) & 0xF);
    const bool diag = (ti == tj);                 // uniform branch

    const int iaXS = (ti * 16 + lx) * XS;         // A-matrix row base (M = lx)
    const int jbXS = (tj * 16 + lx) * XS;         // B-matrix col base (N = lx)
    const int jrow = tj * 16 + lx;
    const int jrb  = jrow * NWIN;
    float* outb = out + (size_t)b * ((size_t)NPAIR * NWIN);

    // Hoist per-(lane,v) output pointers, rinv bases, validity out of the t loop.
    float* ptrs[8];
    int    irb[8];
    int    ival[8];
    #pragma unroll
    for (int v = 0; v < 8; ++v) {
        const int i = ti * 16 + v + 8 * hi;
        ival[v] = i;
        irb[v]  = i * NWIN;
        const int p = (i <= jrow) ? (i * (129 - i)) / 2 + (jrow - i) : 0;
        ptrs[v] = outb + (size_t)p * NWIN;
    }

    for (int t = 0; t < NWIN; ++t) {
        v8f c = {};
        #pragma unroll
        for (int s = 0; s < 3; ++s) {
            int col = t + 4 * s + 2 * hi;          // this lane's low K column
            if (s == 2) col = hi ? 30 : col;       // k=10,11 -> zero-pad columns 30/31
            v2f a, bm;
            a.x  = xn[iaXS + col];
            a.y  = xn[iaXS + col + 1];
            bm.x = xn[jbXS + col];
            bm.y = xn[jbXS + col + 1];
            // D = A(16x4) * B(4x16) + C, f32 accumulate
            c = __builtin_amdgcn_wmma_f32_16x16x4_f32(
                    false, a, false, bm, (short)0, c, false, false);
        }
        const float rj = rinv[jrb + t] * (1.0f / WND);   // fold 1/W into scale
        if (diag) {
            #pragma unroll
            for (int v = 0; v < 8; ++v)
                if (ival[v] <= jrow)
                    ptrs[v][t] = c[v] * rj * rinv[irb[v] + t];
        } else {
            #pragma unroll
            for (int v = 0; v < 8; ++v)
                ptrs[v][t] = c[v] * rj * rinv[irb[v] + t];
        }
    }
}

extern "C" void kernel_launch(void* const* d_in, const int* in_sizes, int n_in,
                              void* d_out, int out_size, void* d_ws, size_t ws_size,
                              hipStream_t stream) {
    const float* x = (const float*)d_in[0];
    float* out = (float*)d_out;
    ts_corr_kernel<<<NB, THREADS, 0, stream>>>(x, out);
}